// PostProcess_77249281786083
// MI455X (gfx1250) — compile-verified
//
#include <hip/hip_runtime.h>
#include <stdint.h>

// Problem constants (from reference)
#define B_      64
#define Q_      10000
#define C_      80
#define QC_     800000          // Q*C per batch
#define TOPK_   100
#define IOU_THR_ 0.7f

// Algorithm constants
#define NBIN_   4096            // histogram bins (top 12 bits of monotonic u32)
#define CAP_    6144            // candidate capacity per batch
#define HTILE_  4096            // floats per async LDS tile (16 KB)
#define HBLOCKS_ 8              // blocks per batch for histogram pass
#define HELEMS_ 100000          // elements per histogram block
#define HTILES_ 25              // ceil(100000/4096)
#define CTILE_  8192            // elements per compaction block
#define CBLOCKS_ 98             // ceil(800000/8192)
#define NPAD_   112             // 100 padded to 7 WMMA tiles of 16
#define NTILE_  7               // tiles per dim

typedef __attribute__((ext_vector_type(2))) float v2f;
typedef __attribute__((ext_vector_type(8))) float v8f;

__device__ __constant__ int d_coco[80] = {
  1,2,3,4,5,6,7,8,9,10,11,13,14,15,16,17,18,19,20,21,22,23,24,25,27,28,31,32,
  33,34,35,36,37,38,39,40,41,42,43,44,46,47,48,49,50,51,52,53,54,55,56,57,58,
  59,60,61,62,63,64,65,67,70,72,73,74,75,76,77,78,79,80,81,82,84,85,86,87,88,89,90
};

// Order-preserving float -> uint32 map (works for all finite floats)
__device__ __forceinline__ uint32_t mono_u32(float f) {
  uint32_t u = __float_as_uint(f);
  return (u & 0x80000000u) ? ~u : (u | 0x80000000u);
}

__device__ __forceinline__ float neg_inf() { return __uint_as_float(0xff800000u); }

// ---- CDNA5 async global->LDS copy (ASYNCcnt path) ----
template<int N>
__device__ __forceinline__ void wait_asynccnt() {
  asm volatile("s_wait_asynccnt %0" :: "n"(N) : "memory");
}

// GVS mode: copies 16 bytes per lane from global address saddr + 32-bit offset vaddr to the LDS address in vdst
__device__ __forceinline__ void async_b128(unsigned lds_off, unsigned src_off,
                                           const float* saddr) {
  asm volatile("global_load_async_to_lds_b128 %0, %1, %2"
               :: "v"(lds_off), "v"(src_off), "s"(saddr) : "memory");
}

// Issue one HTILE_ (4096 floats): 4 x b128 per thread, coalesced.
__device__ __forceinline__ void issue_tile(const float* src, uint32_t base,
                                           int tid, unsigned lds_base) {
#pragma unroll
  for (int g = 0; g < 4; ++g) {
    uint32_t e  = base + (uint32_t)(g * 1024 + tid * 4);
    uint32_t ec = e > (uint32_t)(QC_ - 4) ? (uint32_t)(QC_ - 4) : e; // clamp: stay in batch
    async_b128(lds_base + (unsigned)(g * 1024 + tid * 4) * 4u, ec * 4u, src);
  }
}

// ---------------- Kernel 0: zero histogram + counters ----------------
__global__ void k_zero(uint32_t* p, int n) {
  int i = blockIdx.x * 256 + threadIdx.x;
  if (i < n) p[i] = 0u;
}

// ---------------- Kernel 1: per-batch histogram of logits ----------------
// Streams tiles through LDS via the async copy engine, double buffered.
__global__ void k_hist(const float* __restrict__ logits, uint32_t* __restrict__ hist) {
  __shared__ float    buf[2][HTILE_];
  __shared__ uint32_t lh[NBIN_];
  const int b   = blockIdx.y;
  const int tid = threadIdx.x;
  for (int i = tid; i < NBIN_; i += 256) lh[i] = 0u;

  const float*   src = logits + (size_t)b * QC_;
  const uint32_t s0  = (uint32_t)blockIdx.x * HELEMS_;
  const uint32_t lim = s0 + HELEMS_;             // block-private range (<= QC_)
  const unsigned l0  = (unsigned)(uintptr_t)&buf[0][0];
  const unsigned l1  = (unsigned)(uintptr_t)&buf[1][0];

  issue_tile(src, s0, tid, l0);
  __syncthreads();                               // lh zeroed before any atomic

  for (int k = 0; k < HTILES_; ++k) {
    if (k + 1 < HTILES_) {
      issue_tile(src, s0 + (uint32_t)(k + 1) * HTILE_, tid, ((k + 1) & 1) ? l1 : l0);
      wait_asynccnt<4>();                        // previous tile's 4 ops done
    } else {
      wait_asynccnt<0>();
    }
    const float* bp   = buf[k & 1];
    const uint32_t tb = s0 + (uint32_t)k * HTILE_;
#pragma unroll
    for (int g = 0; g < 4; ++g) {
      const int li = g * 1024 + tid * 4;
      float4 v = *(const float4*)&bp[li];        // ds_load_b128, own data only
      uint32_t e = tb + (uint32_t)li;
      if (e + 0 < lim) atomicAdd(&lh[mono_u32(v.x) >> 20], 1u);
      if (e + 1 < lim) atomicAdd(&lh[mono_u32(v.y) >> 20], 1u);
      if (e + 2 < lim) atomicAdd(&lh[mono_u32(v.z) >> 20], 1u);
      if (e + 3 < lim) atomicAdd(&lh[mono_u32(v.w) >> 20], 1u);
    }
  }
  __syncthreads();
  uint32_t* gh = hist + (size_t)b * NBIN_;
  for (int i = tid; i < NBIN_; i += 256) {
    uint32_t c = lh[i];
    if (c) atomicAdd(&gh[i], c);
  }
}

// ---------------- Kernel 2: find per-batch threshold bin ----------------
__global__ void k_thresh(const uint32_t* __restrict__ hist, uint32_t* __restrict__ uthr) {
  __shared__ uint32_t h[NBIN_];
  const int b = blockIdx.x;
  for (int i = threadIdx.x; i < NBIN_; i += 256) h[i] = hist[(size_t)b * NBIN_ + i];
  __syncthreads();
  if (threadIdx.x == 0) {
    uint32_t acc = 0; int t = 0;
    for (int i = NBIN_ - 1; i >= 0; --i) {
      acc += h[i];
      if (acc >= TOPK_) { t = i; break; }
    }
    uthr[b] = (uint32_t)t << 20;                 // elements with mono >= this are candidates
  }
}

// ---------------- Kernel 3: compact candidates (2nd pass, mostly L2 hits) ----------------
__global__ void k_compact(const float* __restrict__ logits,
                          const uint32_t* __restrict__ uthr,
                          uint32_t* __restrict__ cnt,
                          float* __restrict__ cval, uint32_t* __restrict__ cidx) {
  const int b = blockIdx.y;
  const float* src = logits + (size_t)b * QC_;
  const uint32_t thr  = uthr[b];
  const uint32_t base = (uint32_t)blockIdx.x * CTILE_;
#pragma unroll
  for (int g = 0; g < 8; ++g) {
    uint32_t e = base + (uint32_t)(g * 1024 + threadIdx.x * 4);
    if (e + 3 < QC_) {
      float4 v = *(const float4*)&src[e];
      float vv[4] = {v.x, v.y, v.z, v.w};
#pragma unroll
      for (int j = 0; j < 4; ++j) {
        if (mono_u32(vv[j]) >= thr) {
          uint32_t p = atomicAdd(&cnt[b], 1u);
          if (p < CAP_) { cval[(size_t)b * CAP_ + p] = vv[j]; cidx[(size_t)b * CAP_ + p] = e + j; }
        }
      }
    } else {
      for (int j = 0; j < 4; ++j) {
        if (e + j < QC_) {
          float f = src[e + j];
          if (mono_u32(f) >= thr) {
            uint32_t p = atomicAdd(&cnt[b], 1u);
            if (p < CAP_) { cval[(size_t)b * CAP_ + p] = f; cidx[(size_t)b * CAP_ + p] = e + j; }
          }
        }
      }
    }
  }
}

// ---------------- Kernel 4: exact top-100 + box gather/scale + partial outputs ----------------
__global__ void k_select(const float* __restrict__ cval, const uint32_t* __restrict__ cidx,
                         const uint32_t* __restrict__ cnt,
                         const float* __restrict__ pred_boxes,
                         const float* __restrict__ tsizes,
                         float* __restrict__ out, int* __restrict__ scls) {
  __shared__ float    sv[CAP_];
  __shared__ uint32_t si[CAP_];
  __shared__ float    redv[256];
  __shared__ uint32_t redi[256];
  __shared__ uint32_t redp[256];
  __shared__ float    topv[TOPK_];
  __shared__ uint32_t topi[TOPK_];
  const int b = blockIdx.x, tid = threadIdx.x;
  const uint32_t n = min(cnt[b], (uint32_t)CAP_);

  for (int j = tid; j < CAP_; j += 256) {
    bool ok = (uint32_t)j < n;
    sv[j] = ok ? cval[(size_t)b * CAP_ + j] : neg_inf();
    si[j] = ok ? cidx[(size_t)b * CAP_ + j] : 0xFFFFFFFFu;
  }
  __syncthreads();

  for (int it = 0; it < TOPK_; ++it) {
    float bv = neg_inf(); uint32_t bi = 0xFFFFFFFFu, bp = 0xFFFFFFFFu;
    for (int j = tid; j < CAP_; j += 256) {
      float v = sv[j]; uint32_t ii = si[j];
      if (v > bv || (v == bv && ii < bi)) { bv = v; bi = ii; bp = (uint32_t)j; }
    }
    redv[tid] = bv; redi[tid] = bi; redp[tid] = bp;
    __syncthreads();
    for (int s2 = 128; s2 > 0; s2 >>= 1) {
      if (tid < s2) {
        float v2 = redv[tid + s2]; uint32_t i2 = redi[tid + s2];
        if (v2 > redv[tid] || (v2 == redv[tid] && i2 < redi[tid])) {
          redv[tid] = v2; redi[tid] = i2; redp[tid] = redp[tid + s2];
        }
      }
      __syncthreads();
    }
    if (tid == 0) {
      topv[it] = redv[0]; topi[it] = redi[0];
      if (redp[0] < CAP_) sv[redp[0]] = neg_inf();
    }
    __syncthreads();
  }

  if (tid < TOPK_) {
    const float lv = topv[tid];
    const uint32_t fi = topi[tid];
    const uint32_t q  = fi / C_;
    const uint32_t cl = fi % C_;
    const float score = 1.0f / (1.0f + expf(-lv));
    const float* pb = pred_boxes + ((size_t)b * Q_ + q) * 4;
    const float cx = pb[0], cy = pb[1], w = pb[2], h = pb[3];
    const float ih = tsizes[2 * b], iw = tsizes[2 * b + 1];
    out[(size_t)b * TOPK_ + tid]                    = score;
    out[(size_t)B_ * TOPK_ + b * TOPK_ + tid]       = (float)d_coco[cl];
    float* ob = out + 2 * (size_t)B_ * TOPK_ + ((size_t)b * TOPK_ + tid) * 4;
    ob[0] = (cx - 0.5f * w) * iw;
    ob[1] = (cy - 0.5f * h) * ih;
    ob[2] = (cx + 0.5f * w) * iw;
    ob[3] = (cy + 0.5f * h) * ih;
    scls[b * TOPK_ + tid] = (int)cl;
  }
}

// ---------------- Kernel 5: class-aware NMS, full IoU matrix like the reference ----
// Pairwise area-sum matrix S[i][j] = area[i] + area[j] comes from the matrix pipe:
// V_WMMA_F32_16X16X4_F32 with A = [area | 1 | 0 | 0] (16x4), B = [1 ; area ; 0 ; 0] (4x16).
// Intersection (min/max) on VALU, suppression bits OR'd into an LDS bitmask, then the
// exact greedy propagation (sup = iou>thr & j>i & keep[i]) on 4-word masks.
__global__ void k_nms(const float* __restrict__ boxes_in, const int* __restrict__ scls,
                      float* __restrict__ out_keep) {
  __shared__ float    bo[NPAD_ * 4];   // offset boxes (padded rows zeroed)
  __shared__ float    area[NPAD_];
  __shared__ int      cls[TOPK_];
  __shared__ uint32_t sup[NPAD_][4];   // sup[i] = bitmask of j with iou(i,j) > thr
  __shared__ float    red[128];
  __shared__ uint32_t keepw[4];
  const int b = blockIdx.x, tid = threadIdx.x;      // 128 threads = 4 waves
  const int lane = tid & 31, wave = tid >> 5;
  const float* src = boxes_in + (size_t)b * TOPK_ * 4;

  for (int j = tid; j < TOPK_ * 4; j += 128) bo[j] = src[j];
  for (int j = TOPK_ * 4 + tid; j < NPAD_ * 4; j += 128) bo[j] = 0.0f;
  if (tid < TOPK_) cls[tid] = scls[b * TOPK_ + tid];
  for (int j = tid; j < NPAD_ * 4; j += 128) sup[0][j] = 0u;  // flat clear
  __syncthreads();

  // max coordinate over the 100x4 scaled boxes (reference: max_coord = max + 1)
  float m = neg_inf();
  for (int j = tid; j < TOPK_ * 4; j += 128) m = fmaxf(m, bo[j]);
  red[tid] = m;
  for (int s2 = 64; s2 > 0; s2 >>= 1) {
    __syncthreads();
    if (tid < s2) red[tid] = fmaxf(red[tid], red[tid + s2]);
  }
  __syncthreads();
  const float maxc = red[0] + 1.0f;

  if (tid < TOPK_) {
    const float o = (float)cls[tid] * maxc;      // batched-NMS class offset
    bo[tid * 4 + 0] += o; bo[tid * 4 + 1] += o;
    bo[tid * 4 + 2] += o; bo[tid * 4 + 3] += o;
  }
  __syncthreads();
  if (tid < NPAD_)
    area[tid] = (tid < TOPK_)
      ? (bo[tid * 4 + 2] - bo[tid * 4 + 0]) * (bo[tid * 4 + 3] - bo[tid * 4 + 1])
      : 0.0f;
  __syncthreads();

  // 7x7 tiles of 16x16; one tile per wave per round (loop bounds are wave-uniform,
  // so EXEC is all ones at every WMMA).
  for (int t = wave; t < NTILE_ * NTILE_; t += 4) {
    const int ti = t / NTILE_, tj = t % NTILE_;
    const int Mb = ti * 16, Nb = tj * 16;
    v2f a, bb;
    a[0]  = (lane < 16) ? area[Mb + lane] : 0.0f;  // A: K=0 col = area[M] (K=2 lanes zero)
    a[1]  = (lane < 16) ? 1.0f            : 0.0f;  //    K=1 col = 1       (K=3 lanes zero)
    bb[0] = (lane < 16) ? 1.0f            : 0.0f;  // B: K=0 row = 1       (K=2 lanes zero)
    bb[1] = (lane < 16) ? area[Nb + lane] : 0.0f;  //    K=1 row = area[N] (K=3 lanes zero)
    v8f c = {};
    v8f d = __builtin_amdgcn_wmma_f32_16x16x4_f32(
        false, a, false, bb, (short)0, c, false, false);
    // D layout: VGPR r holds M = Mb + r + (lane<16 ? 0 : 8), N = Nb + (lane&15)
    const int N = Nb + (lane & 15);
    const int mhi = (lane < 16) ? 0 : 8;
#pragma unroll
    for (int r = 0; r < 8; ++r) {
      const int M = Mb + r + mhi;
      if (M < TOPK_ && N < TOPK_) {
        const float lx = fmaxf(bo[M * 4 + 0], bo[N * 4 + 0]);
        const float ly = fmaxf(bo[M * 4 + 1], bo[N * 4 + 1]);
        const float rx = fminf(bo[M * 4 + 2], bo[N * 4 + 2]);
        const float ry = fminf(bo[M * 4 + 3], bo[N * 4 + 3]);
        const float iw = fmaxf(rx - lx, 0.0f), ih = fmaxf(ry - ly, 0.0f);
        const float inter = iw * ih;
        const float iou = inter / (d[r] - inter);  // d[r] = area[M] + area[N]
        if (iou > IOU_THR_)
          atomicOr(&sup[M][N >> 5], 1u << (N & 31));
      }
    }
  }
  __syncthreads();

  // Exact greedy propagation on bitmasks (matches reference fori_loop).
  if (tid == 0) {
    uint32_t k0 = 0xFFFFFFFFu, k1 = 0xFFFFFFFFu, k2 = 0xFFFFFFFFu, k3 = 0x0000000Fu;
    uint32_t kw[4] = {k0, k1, k2, k3};
    for (int i = 0; i < TOPK_; ++i) {
      if ((kw[i >> 5] >> (i & 31)) & 1u) {
        const uint32_t wlo = (uint32_t)(i >> 5);
        const uint32_t lowmask = (uint32_t)((2ull << (i & 31)) - 1ull); // bits <= i
#pragma unroll
        for (int w = 0; w < 4; ++w) {
          uint32_t s = sup[i][w];
          if ((uint32_t)w < wlo) s = 0u;
          else if ((uint32_t)w == wlo) s &= ~lowmask;   // only j > i
          kw[w] &= ~s;
        }
      }
    }
    keepw[0] = kw[0]; keepw[1] = kw[1]; keepw[2] = kw[2]; keepw[3] = kw[3];
  }
  __syncthreads();
  if (tid < TOPK_)
    out_keep[b * TOPK_ + tid] = ((keepw[tid >> 5] >> (tid & 31)) & 1u) ? 1.0f : 0.0f;
}

// ---------------- host launcher ----------------
extern "C" void kernel_launch(void* const* d_in, const int* in_sizes, int n_in,
                              void* d_out, int out_size, void* d_ws, size_t ws_size,
                              hipStream_t stream) {
  const float* logits = (const float*)d_in[0];   // [64,10000,80]
  const float* pboxes = (const float*)d_in[1];   // [64,10000,4]
  const float* tsizes = (const float*)d_in[2];   // [64,2]
  float* out = (float*)d_out;                    // scores|labels|boxes|keep = 44800 floats

  // workspace layout (~4.2 MB)
  uint32_t* hist = (uint32_t*)d_ws;              // B*NBIN
  uint32_t* cnt  = hist + (size_t)B_ * NBIN_;    // B
  uint32_t* uthr = cnt + B_;                     // B
  float*    cval = (float*)(uthr + B_);          // B*CAP
  uint32_t* cidx = (uint32_t*)(cval + (size_t)B_ * CAP_); // B*CAP
  int*      scls = (int*)(cidx + (size_t)B_ * CAP_);      // B*TOPK

  const int nz = B_ * NBIN_ + B_;                // hist + cnt are contiguous
  k_zero<<<(nz + 255) / 256, 256, 0, stream>>>(hist, nz);
  k_hist<<<dim3(HBLOCKS_, B_), 256, 0, stream>>>(logits, hist);
  k_thresh<<<B_, 256, 0, stream>>>(hist, uthr);
  k_compact<<<dim3(CBLOCKS_, B_), 256, 0, stream>>>(logits, uthr, cnt, cval, cidx);
  k_select<<<B_, 256, 0, stream>>>(cval, cidx, cnt, pboxes, tsizes, out, scls);
  k_nms<<<B_, 128, 0, stream>>>(out + 2 * (size_t)B_ * TOPK_, scls,
                                out + 2 * (size_t)B_ * TOPK_ + (size_t)B_ * TOPK_ * 4);
  (void)in_sizes; (void)n_in; (void)out_size; (void)ws_size;
}